// AuditoryCaps_13486197309459
// MI455X (gfx1250) — compile-verified
//
#include <hip/hip_runtime.h>
#include <math.h>

// ---------------------------------------------------------------------------
// Capsule dynamic routing, fused for MI455X (gfx1250, wave32, WMMA).
//
//  u_hat is NEVER materialized (151 MB). Agreement logits are linear in v:
//    b_r[b,i,c] = dot(u_hat[b,i,c,:], Vacc[b,c,:]),  Vacc = sum of prior v's
//  so each routing iteration is one sweep over I recomputing u_hat tiles with
//  V_WMMA_F32_16X16X4_F32 (fp32-exact, K=8 -> 2 WMMA per 16x16 tile).
//
//  Tile orientation: D[e, b] = sum_d W[i,c,d,e] * x[b,i,d]
//    A (16x4 f32): A[m=e, k=d]  -> lane l: e=l%16, elem v: d = 4*kstep+v+2*(l/16)
//    B (4x16 f32): B[k=d, n=b]  -> lane l: b=l%16, elem v: d = 4*kstep+v+2*(l/16)
//    D (16x16)   : VGPR j, lane l -> e = j + 8*(l/16), b = l%16
//  => logit over e = 8 reg-FMAs; the two half-wave partials are stored to
//     separate LDS slots (no cross-lane op, no predication) and summed in the
//     softmax phase, which maps softmax-over-C=32 onto one wave32 (c = lane).
// ---------------------------------------------------------------------------

typedef float v2f __attribute__((ext_vector_type(2)));
typedef float v8f __attribute__((ext_vector_type(8)));

#define B_      64
#define I_      1152
#define C_      32
#define E_      16
#define D_      8
#define BT      16            // batch tile = WMMA N
#define NBT     (B_ / BT)     // 4 batch tiles
#define NSPLIT  48            // I-chunks per sweep (grid = 4*48 = 192 blocks)
#define IPER    (I_ / NSPLIT) // 24 i's per workgroup
#define SVE     (B_ * C_ * E_) // 32768 floats: s / Vacc / out element count

__global__ __launch_bounds__(256)
void caps_sweep(const float* __restrict__ x, const float* __restrict__ W,
                const float* __restrict__ Vacc, float* __restrict__ s_glob)
{
    __shared__ float lg[2][BT][C_ + 1]; // logit half-partials [half][b][c]
    __shared__ float cf[BT][C_ + 1];    // softmax coef[b_local][c]

    const int tid   = threadIdx.x;
    const int w     = tid >> 5;        // wave 0..7; owns c = 4w..4w+3
    const int lane  = tid & 31;
    const int half  = lane >> 4;       // 0/1
    const int bl    = lane & 15;       // local batch (N index of WMMA tile)
    const int btile = blockIdx.x & (NBT - 1);
    const int chunk = blockIdx.x / NBT;
    const int i0    = chunk * IPER;
    const int bg    = btile * BT + bl; // global batch for this lane

    // Per-sweep constants: Vacc operand in D-tile layout, and s partials.
    float vac[4][8];
    float sacc[4][8];
    for (int cc = 0; cc < 4; ++cc) {
        const int c = (w << 2) + cc;
        for (int j = 0; j < 8; ++j) {
            const int e = j + (half << 3);
            vac[cc][j]  = Vacc[(bg * C_ + c) * E_ + e];
            sacc[cc][j] = 0.0f;
        }
    }

    v8f dt[4];  // u_hat tiles for this wave's 4 capsules (current i)

    for (int ii = 0; ii < IPER; ++ii) {
        const int i = i0 + ii;

        // Prefetch next i's W row for this wave's capsules (overlaps WMMA).
        if (ii + 1 < IPER) {
            const float* Wn = W + (((size_t)(i + 1) * C_ + (w << 2)) * D_) * E_;
            __builtin_prefetch(Wn + lane, 0, 1);      // global_prefetch_b8
        }

        // --- B operand (x), shared across the wave's 4 capsules -----------
        v2f bv0, bv1;
        {
            const float* xp = x + ((size_t)bg * I_ + i) * D_;
            const int d0 = half << 1;              // 0 or 2
            bv0[0] = xp[d0 + 0]; bv0[1] = xp[d0 + 1];         // k-step 0: d 0..3
            bv1[0] = xp[4 + d0 + 0]; bv1[1] = xp[4 + d0 + 1]; // k-step 1: d 4..7
        }

        // --- Phase A: WMMA u_hat tiles + logit half-partials --------------
        for (int cc = 0; cc < 4; ++cc) {
            const int c = (w << 2) + cc;
            const float* Wp = W + (((size_t)i * C_ + c) * D_) * E_;
            const int d0 = half << 1;
            v2f a0, a1;
            a0[0] = Wp[(d0 + 0) * E_ + bl];
            a0[1] = Wp[(d0 + 1) * E_ + bl];
            a1[0] = Wp[(4 + d0 + 0) * E_ + bl];
            a1[1] = Wp[(4 + d0 + 1) * E_ + bl];

            v8f d = {};
            d = __builtin_amdgcn_wmma_f32_16x16x4_f32(false, a0, false, bv0,
                                                      (short)0, d, false, false);
            d = __builtin_amdgcn_wmma_f32_16x16x4_f32(false, a1, false, bv1,
                                                      (short)0, d, false, false);
            dt[cc] = d;

            // half-partial of logit[bg, c] = sum_e u_hat * Vacc
            float t = 0.0f;
            for (int j = 0; j < 8; ++j) t = fmaf(d[j], vac[cc][j], t);
            lg[half][bl][c] = t;   // unconditional, conflict-free (stride 33)
        }
        __syncthreads();

        // --- Phase B: softmax over C=32 (one wave = one b-row, c = lane) --
        for (int rr = 0; rr < 2; ++rr) {
            const int rb = (w << 1) + rr;        // local batch row
            const float v = lg[0][rb][lane] + lg[1][rb][lane];
            float m = v;
            for (int mk = 16; mk >= 1; mk >>= 1)
                m = fmaxf(m, __shfl_xor(m, mk, 32));
            const float ex = __expf(v - m);
            float z = ex;
            for (int mk = 16; mk >= 1; mk >>= 1)
                z += __shfl_xor(z, mk, 32);
            cf[rb][lane] = ex / z;
        }
        __syncthreads();

        // --- Phase C: s[b,c,e] += coef * u_hat ----------------------------
        for (int cc = 0; cc < 4; ++cc) {
            const int c = (w << 2) + cc;
            const float coef = cf[bl][c];
            for (int j = 0; j < 8; ++j)
                sacc[cc][j] = fmaf(coef, dt[cc][j], sacc[cc][j]);
        }
        // next iteration's Phase-A lg writes are ordered by the two barriers
        // above (Phase C only reads cf, Phase A only writes lg).
    }

    // Cross-chunk reduction of s partials (48 adders per location).
    for (int cc = 0; cc < 4; ++cc) {
        const int c = (w << 2) + cc;
        for (int j = 0; j < 8; ++j) {
            const int e = j + (half << 3);
            atomicAdd(&s_glob[(bg * C_ + c) * E_ + e], sacc[cc][j]);
        }
    }
}

// squash s -> v, fold v into Vacc, write v to out, reset s for next sweep.
__global__ __launch_bounds__(256)
void caps_squash(float* __restrict__ s, float* __restrict__ Vacc,
                 float* __restrict__ out)
{
    const int t = blockIdx.x * 256 + threadIdx.x;   // 0..SVE-1, e = t%16
    const float val = s[t];
    float sq = val * val;
    sq += __shfl_xor(sq, 1, 32);
    sq += __shfl_xor(sq, 2, 32);
    sq += __shfl_xor(sq, 4, 32);
    sq += __shfl_xor(sq, 8, 32);
    const float scale = sq / (1.0f + sq) / (sqrtf(sq) + 1e-8f);
    const float v = scale * val;
    out[t]  = v;
    Vacc[t] += v;
    s[t]    = 0.0f;   // ready for next sweep
}

__global__ __launch_bounds__(256)
void caps_init(float* __restrict__ ws)
{
    const int t = blockIdx.x * 256 + threadIdx.x;
    ws[t] = 0.0f;     // zero Vacc and s (2*SVE floats)
}

extern "C" void kernel_launch(void* const* d_in, const int* in_sizes, int n_in,
                              void* d_out, int out_size, void* d_ws, size_t ws_size,
                              hipStream_t stream)
{
    (void)in_sizes; (void)n_in; (void)out_size; (void)ws_size;
    const float* x = (const float*)d_in[0];   // [64,1152,8]
    const float* W = (const float*)d_in[1];   // [1152,32,8,16]
    float* out  = (float*)d_out;              // [64,32,16]
    float* Vacc = (float*)d_ws;               // SVE floats
    float* s    = Vacc + SVE;                 // SVE floats  (total 256 KB)

    caps_init<<<(2 * SVE) / 256, 256, 0, stream>>>(Vacc);
    for (int r = 0; r < 3; ++r) {
        caps_sweep <<<NBT * NSPLIT, 256, 0, stream>>>(x, W, Vacc, s);
        caps_squash<<<SVE / 256,   256, 0, stream>>>(s, Vacc, out);
    }
}